// GraphConvolution_37958920962295
// MI455X (gfx1250) — compile-verified
//
#include <hip/hip_runtime.h>
#include <hip/hip_bf16.h>

typedef __attribute__((ext_vector_type(2))) float v2f;
typedef __attribute__((ext_vector_type(8))) float v8f;

#define F_IN   128
#define K_TOT  256          // 2*F_IN
#define F_OUT  128
#define LDS_STRIDE 260      // 16 rows x 260 floats; 260 % 64 = 4 -> spread banks, even -> v2f aligned

// ---------------- zero workspace ----------------
__global__ void gc_zero_kernel(float* __restrict__ p, long long n) {
    long long i = (long long)blockIdx.x * blockDim.x + threadIdx.x;
    if (i < n) p[i] = 0.0f;
}

// ---------------- SpMM scatter: h[src] += val * x[dst] ----------------
// 32 lanes per edge, each lane handles 4 consecutive features (float4 gather).
__global__ __launch_bounds__(256) void gc_scatter_kernel(
    const float* __restrict__ x, const int* __restrict__ esrc,
    const int* __restrict__ edst, const float* __restrict__ eval,
    float* __restrict__ h, long long E) {
    long long tid = (long long)blockIdx.x * blockDim.x + threadIdx.x;
    long long total = E << 5;                 // E * 32
    if (tid >= total) return;
    long long e = tid >> 5;
    int lane = (int)(tid & 31);
    int s = esrc[e];
    int d = edst[e];
    float v = eval[e];
    const float4* xrow = (const float4*)(x + (long long)d * F_IN);
    float4 xv = xrow[lane];                   // coalesced: wave == one edge row
    float* hp = h + (long long)s * F_IN + lane * 4;
    atomicAdd(hp + 0, v * xv.x);
    atomicAdd(hp + 1, v * xv.y);
    atomicAdd(hp + 2, v * xv.z);
    atomicAdd(hp + 3, v * xv.w);
}

// ---------------- rowsum[src] += val ----------------
__global__ void gc_rowsum_kernel(const int* __restrict__ esrc,
                                 const float* __restrict__ eval,
                                 float* __restrict__ rowsum, long long E) {
    long long e = (long long)blockIdx.x * blockDim.x + threadIdx.x;
    if (e < E) atomicAdd(&rowsum[esrc[e]], eval[e]);
}

// ---------------- GEMM: out = [h | ori] @ W, /rowsum, +bias ----------------
// Block = 256 threads = 8 waves. Block handles 16 output rows; wave w handles
// output columns [16w, 16w+16). Inner loop: 64 x v_wmma_f32_16x16x4_f32.
__global__ __launch_bounds__(256) void gc_gemm_kernel(
    const float* __restrict__ h, const float* __restrict__ ori,
    const float* __restrict__ W, const float* __restrict__ bias,
    const float* __restrict__ rowsum, float* __restrict__ out, int N) {
    __shared__ float Atile[16 * LDS_STRIDE];

    const int base = blockIdx.x * 16;        // first output row of this block
    const int tid  = threadIdx.x;

    // Stage A panel: 16 rows x 256 K (cols 0..127 from h, 128..255 from ori).
    // 4096 floats = 1024 float4 loads across 256 threads (4 each).
    #pragma unroll
    for (int i = 0; i < 4; ++i) {
        int idx = tid + i * 256;             // float4 index, 64 float4 per row
        int r   = idx >> 6;
        int col = (idx & 63) * 4;
        int row = base + r; if (row >= N) row = N - 1;   // clamp (N%16==0 normally)
        const float* sp = (col < F_IN)
            ? (h   + (long long)row * F_IN + col)
            : (ori + (long long)row * F_IN + (col - F_IN));
        float4 v = *(const float4*)sp;
        float* dp = &Atile[r * LDS_STRIDE + col];
        dp[0] = v.x; dp[1] = v.y; dp[2] = v.z; dp[3] = v.w;
    }
    __syncthreads();

    const int wave    = tid >> 5;            // 0..7 -> column tile
    const int lane    = tid & 31;
    const int lmod    = lane & 15;
    const int lhi     = lane >> 4;           // 0 or 1 (selects K pair 0..1 vs 2..3)
    const int colBase = wave * 16;
    const int col     = colBase + lmod;

    v8f acc = {};
    #pragma unroll
    for (int k = 0; k < K_TOT; k += 4) {
        // A fragment (16x4 fp32): lane l<16 -> A[l][k..k+1], lane l>=16 -> A[l-16][k+2..k+3]
        v2f a = *(const v2f*)&Atile[lmod * LDS_STRIDE + k + 2 * lhi];
        // B fragment (4x16 fp32): lane l<16 -> W[k][col],W[k+1][col]; l>=16 -> W[k+2],W[k+3]
        int kb = k + 2 * lhi;
        v2f b;
        b.x = W[(long long)kb * F_OUT + col];
        b.y = W[(long long)(kb + 1) * F_OUT + col];
        acc = __builtin_amdgcn_wmma_f32_16x16x4_f32(
            /*neg_a=*/false, a, /*neg_b=*/false, b,
            /*c_mod=*/(short)0, acc, /*reuse_a=*/false, /*reuse_b=*/false);
    }

    // D layout: VGPR r, lanes 0-15 -> row base+r; lanes 16-31 -> row base+8+r.
    const float bv = bias[col];
    #pragma unroll
    for (int r = 0; r < 8; ++r) {
        int row = base + r + 8 * lhi;
        if (row < N)
            out[(long long)row * F_OUT + col] = acc[r] / rowsum[row] + bv;
    }
}

extern "C" void kernel_launch(void* const* d_in, const int* in_sizes, int n_in,
                              void* d_out, int out_size, void* d_ws, size_t ws_size,
                              hipStream_t stream) {
    const float* ori  = (const float*)d_in[0];
    const float* x    = (const float*)d_in[1];
    const int*   esrc = (const int*)d_in[2];
    const int*   edst = (const int*)d_in[3];
    const float* eval = (const float*)d_in[4];
    const float* W    = (const float*)d_in[5];
    const float* bias = (const float*)d_in[6];
    float* out = (float*)d_out;

    const long long N = (long long)in_sizes[0] / F_IN;
    const long long E = (long long)in_sizes[2];

    float* h      = (float*)d_ws;            // N*128 floats
    float* rowsum = h + N * F_IN;            // N floats

    // 1) zero h + rowsum (atomics accumulate into them)
    long long zn = N * (F_IN + 1);
    gc_zero_kernel<<<(unsigned)((zn + 255) / 256), 256, 0, stream>>>(h, zn);

    // 2) scatter SpMM + rowsum
    long long st = E << 5;
    gc_scatter_kernel<<<(unsigned)((st + 255) / 256), 256, 0, stream>>>(
        x, esrc, edst, eval, h, E);
    gc_rowsum_kernel<<<(unsigned)((E + 255) / 256), 256, 0, stream>>>(
        esrc, eval, rowsum, E);

    // 3) fused WMMA GEMM + mean divide + bias
    gc_gemm_kernel<<<(unsigned)((N + 15) / 16), 256, 0, stream>>>(
        h, ori, W, bias, rowsum, out, (int)N);
}